// QuLinear_17454747091654
// MI455X (gfx1250) — compile-verified
//
#include <hip/hip_runtime.h>

// Quantum circuit simulator (12 qubits, batch 1024) for MI455X / gfx1250.
//   out[b, j] = sum_{k<16} |amp[j*16+k]|^2 after
//   encode(Ry) -> YZY -> ring-CNOT -> YZY
// Strategy:
//   * encoding + first YZY layer: product state (12 complex 2-vectors)
//   * ring CNOT: fixed bit-permutation folded into state materialization
//   * second YZY layer: A(16x16) (x) B(16x16) (x) C(16x16) complex kron ->
//     three 16x16 x 16x256 complex GEMMs executed with V_WMMA_F32_16X16X4_F32
// Everything lives in LDS; one block per batch element; wave32, 8 waves/block.

#define NQ   12
#define DIM  4096
#define N3   36

typedef float v2f __attribute__((ext_vector_type(2)));
typedef float v8f __attribute__((ext_vector_type(8)));

__device__ __forceinline__ v8f wmma4(v2f a, v2f b, v8f c) {
  // D = A(16x4) * B(4x16) + C, fp32 in/out. f32 has no A/B neg -> pass false.
  return __builtin_amdgcn_wmma_f32_16x16x4_f32(false, a, false, b,
                                               (short)0, c, false, false);
}

// One axis contraction of the (16,16,16) state with a 16x16 complex gate G.
// MODE 0: new[a,b,c] = sum_a' G[a,a'] S[a',b,c]   (chunk j = b)
// MODE 1: new[a,b,c] = sum_b' G[b,b'] S[a,b',c]   (chunk j = a)
// MODE 2: new[a,b,c] = sum_c' G[c,c'] S[a,b,c']   (chunk j = a)
// Each wave owns chunks {2w, 2w+1}; input region == output region per chunk,
// and chunks are disjoint across waves -> in-place, no intra-phase barrier.
template <int MODE>
__device__ __forceinline__ void contract(float* Sre, float* Sim,
                                         const float* Gr, const float* Gi,
                                         int wave, int lane) {
  const int hi = (lane >> 4) & 1;  // half-wave select
  const int lh = lane & 15;
#pragma unroll
  for (int rep = 0; rep < 2; ++rep) {
    const int j = wave * 2 + rep;
    float ar[8], ai[8], br[8], bi[8];
#pragma unroll
    for (int kc = 0; kc < 4; ++kc) {
#pragma unroll
      for (int v = 0; v < 2; ++v) {
        const int k = kc * 4 + v + hi * 2;  // K index this (vgpr, half) holds
        // A operand: gate row = lh, col = k (row-major 16x16 in LDS)
        ar[kc * 2 + v] = Gr[lh * 16 + k];
        ai[kc * 2 + v] = Gi[lh * 16 + k];
        // B operand: state element S[k, n=lh] of this chunk
        const int idx = (MODE == 0) ? (k * 256 + j * 16 + lh)
                      : (MODE == 1) ? (j * 256 + k * 16 + lh)
                                    : (j * 256 + lh * 16 + k);
        br[kc * 2 + v] = Sre[idx];
        bi[kc * 2 + v] = Sim[idx];
      }
    }
    v8f accR = {0.f, 0.f, 0.f, 0.f, 0.f, 0.f, 0.f, 0.f};
    v8f accI = {0.f, 0.f, 0.f, 0.f, 0.f, 0.f, 0.f, 0.f};
#pragma unroll
    for (int kc = 0; kc < 4; ++kc) {
      v2f Ar = {ar[2 * kc], ar[2 * kc + 1]};
      v2f Ai = {ai[2 * kc], ai[2 * kc + 1]};
      v2f An = {-ai[2 * kc], -ai[2 * kc + 1]};  // -Gi for the real part
      v2f Br = {br[2 * kc], br[2 * kc + 1]};
      v2f Bi = {bi[2 * kc], bi[2 * kc + 1]};
      accR = wmma4(Ar, Br, accR);  // + Gr*Sr
      accR = wmma4(An, Bi, accR);  // - Gi*Si
      accI = wmma4(Ai, Br, accI);  // + Gi*Sr
      accI = wmma4(Ar, Bi, accI);  // + Gr*Si
    }
#pragma unroll
    for (int jd = 0; jd < 8; ++jd) {  // D layout: vgpr jd -> M = jd + hi*8
      const int m = jd + hi * 8;
      const int idx = (MODE == 0) ? (m * 256 + j * 16 + lh)
                    : (MODE == 1) ? (j * 256 + m * 16 + lh)
                                  : (j * 256 + lh * 16 + m);
      Sre[idx] = accR[jd];
      Sim[idx] = accI[jd];
    }
  }
}

__global__ __launch_bounds__(256) void qsim_kernel(const float* __restrict__ x,
                                                   const float* __restrict__ w,
                                                   float* __restrict__ out) {
  __shared__ float Sre[DIM], Sim[DIM];          // state vector
  __shared__ float Gr[3][256], Gi[3][256];      // 16x16 kron gates (layer 1)
  __shared__ float U0r[NQ][2][2], U0i[NQ][2][2];
  __shared__ float U1r[NQ][2][2], U1i[NQ][2][2];
  __shared__ float vr2[NQ][2], vi2[NQ][2];      // per-qubit product-state vecs

  const int t = threadIdx.x;
  const int b = blockIdx.x;

  // ---- Phase 1: combined U = Ry(c)Rz(bb)Ry(a) per qubit, both layers ----
  if (t < 2 * NQ) {
    const int l = t / NQ, q = t % NQ;
    const float a  = w[l * N3 + 3 * q + 0];
    const float bb = w[l * N3 + 3 * q + 1];
    const float c  = w[l * N3 + 3 * q + 2];
    const float c1 = cosf(0.5f * a),  s1 = sinf(0.5f * a);
    const float pc = cosf(0.5f * bb), ps = sinf(0.5f * bb);
    const float c2 = cosf(0.5f * c),  s2 = sinf(0.5f * c);
    const float cc1 = c2 * c1, ss1 = s2 * s1, cs1 = c2 * s1, sc1 = s2 * c1;
    float (*Ur)[2] = (l == 0) ? U0r[q] : U1r[q];
    float (*Ui)[2] = (l == 0) ? U0i[q] : U1i[q];
    Ur[0][0] =  pc * (cc1 - ss1);  Ui[0][0] = -ps * (cc1 + ss1);  // U00
    Ur[0][1] = -pc * (cs1 + sc1);  Ui[0][1] =  ps * (cs1 - sc1);  // U01
    Ur[1][0] =  pc * (sc1 + cs1);  Ui[1][0] =  ps * (cs1 - sc1);  // U10
    Ur[1][1] =  pc * (cc1 - ss1);  Ui[1][1] =  ps * (cc1 + ss1);  // U11
  }
  __syncthreads();

  // ---- Phase 2: kron 16x16 gates for layer 1; encoded 2-vectors (layer 0) --
  {
    const int r = t >> 4, c = t & 15;
#pragma unroll
    for (int g = 0; g < 3; ++g) {
      float pr = 1.f, pi = 0.f;
#pragma unroll
      for (int jq = 0; jq < 4; ++jq) {
        const int q  = g * 4 + jq;
        const int rb = (r >> (3 - jq)) & 1;
        const int cb = (c >> (3 - jq)) & 1;
        const float fr = U1r[q][rb][cb], fi = U1i[q][rb][cb];
        const float nr = pr * fr - pi * fi;
        const float ni = pr * fi + pi * fr;
        pr = nr; pi = ni;
      }
      Gr[g][t] = pr; Gi[g][t] = pi;
    }
  }
  if (t < NQ) {
    const float ang = 0.5f * x[(size_t)b * DIM + t];
    const float ce = cosf(ang), se = sinf(ang);
    vr2[t][0] = U0r[t][0][0] * ce + U0r[t][0][1] * se;
    vi2[t][0] = U0i[t][0][0] * ce + U0i[t][0][1] * se;
    vr2[t][1] = U0r[t][1][0] * ce + U0r[t][1][1] * se;
    vi2[t][1] = U0i[t][1][0] * ce + U0i[t][1][1] * se;
  }
  __syncthreads();

  // ---- Phase 3: materialize state with ring-CNOT permutation folded in ----
  for (int n = 0; n < 16; ++n) {
    const int i = t * 16 + n;
    int m = i;  // trace output index back through CNOT(11,0) ... CNOT(0,1)
#pragma unroll
    for (int q = NQ - 1; q >= 0; --q) {
      const int cpos = 11 - q;
      const int tpos = 11 - ((q + 1) % NQ);
      m ^= ((m >> cpos) & 1) << tpos;
    }
    float pr = 1.f, pi = 0.f;
#pragma unroll
    for (int q = 0; q < NQ; ++q) {
      const int bit = (m >> (11 - q)) & 1;
      const float fr = vr2[q][bit], fi = vi2[q][bit];
      const float nr = pr * fr - pi * fi;
      const float ni = pr * fi + pi * fr;
      pr = nr; pi = ni;
    }
    Sre[i] = pr; Sim[i] = pi;
  }
  __syncthreads();

  // ---- Phase 4: second YZY layer = three WMMA complex contractions ----
  const int lane = t & 31, wave = t >> 5;
  contract<0>(Sre, Sim, &Gr[0][0], &Gi[0][0], wave, lane);
  __syncthreads();
  contract<1>(Sre, Sim, &Gr[1][0], &Gi[1][0], wave, lane);
  __syncthreads();
  contract<2>(Sre, Sim, &Gr[2][0], &Gi[2][0], wave, lane);
  __syncthreads();

  // ---- Phase 5: marginal over first 8 qubits ----
  float acc = 0.f;
#pragma unroll
  for (int k = 0; k < 16; ++k) {
    const float re = Sre[t * 16 + k];
    const float im = Sim[t * 16 + k];
    acc += re * re + im * im;
  }
  out[(size_t)b * 256 + t] = acc;
}

extern "C" void kernel_launch(void* const* d_in, const int* in_sizes, int n_in,
                              void* d_out, int out_size, void* d_ws, size_t ws_size,
                              hipStream_t stream) {
  (void)n_in; (void)out_size; (void)d_ws; (void)ws_size;
  const float* x = (const float*)d_in[0];
  const float* w = (const float*)d_in[1];
  float* out = (float*)d_out;
  const int batch = in_sizes[0] / DIM;  // 1024
  qsim_kernel<<<dim3(batch), dim3(256), 0, stream>>>(x, w, out);
}